// GCNPolicy_72224170049793
// MI455X (gfx1250) — compile-verified
//
#include <hip/hip_runtime.h>
#include <hip/hip_bf16.h>

typedef __attribute__((ext_vector_type(2))) float v2f;
typedef __attribute__((ext_vector_type(8))) float v8f;

// ---------------------------------------------------------------------------
// Degree / normalization
// ---------------------------------------------------------------------------
__global__ void init_deg_kernel(float* deg, int n) {
    int i = blockIdx.x * blockDim.x + threadIdx.x;
    if (i < n) deg[i] = 1.0f;   // self-loop contributes 1
}

__global__ void count_deg_kernel(const int* __restrict__ dst, float* deg, int e) {
    int i = blockIdx.x * blockDim.x + threadIdx.x;
    if (i < e) atomicAdd(&deg[dst[i]], 1.0f);
}

__global__ void rsqrt_kernel(float* deg, int n) {
    int i = blockIdx.x * blockDim.x + threadIdx.x;
    if (i < n) deg[i] = rsqrtf(deg[i]);   // deg >= 1 always (self loop)
}

// ---------------------------------------------------------------------------
// Skinny GEMM via fp32 WMMA: Y[n_rows,16] = X[n_rows,K] @ W[K,16]
// One wave per 16-row tile; V_WMMA_F32_16X16X4_F32, K stepped by 4.
// A layout (16x4 f32): lanes 0-15 hold K={kk,kk+1}, lanes 16-31 hold K={kk+2,kk+3}
// B layout (4x16 f32): VGPR0 = row kk (lanes 0-15) / row kk+2 (lanes 16-31), VGPR1 = +1
// D layout (16x16 f32): VGPR r -> row r (lanes 0-15) / row r+8 (lanes 16-31)
// ---------------------------------------------------------------------------
__global__ void gemm16_wmma_f32(const float* __restrict__ X,
                                const float* __restrict__ W,
                                float* __restrict__ Y,
                                int n_rows, int K) {
    int wave = (int)((blockIdx.x * blockDim.x + threadIdx.x) >> 5);
    int lane = threadIdx.x & 31;
    int m0   = wave * 16;
    if (m0 >= n_rows) return;                 // whole wave exits together -> EXEC all-1 for WMMA

    int half = lane >> 4;                     // 0: K lo pair, 1: K hi pair
    int l    = lane & 15;                     // row within tile (A) / column (B,D)

    const float* xrow = X + (size_t)(m0 + l) * K;
    v8f acc = {};

    for (int kk = 0; kk < K; kk += 4) {
        v2f a, b;
        int ka = kk + 2 * half;
        a.x = xrow[ka];
        a.y = xrow[ka + 1];
        b.x = W[(size_t)ka * 16 + l];
        b.y = W[(size_t)(ka + 1) * 16 + l];
        acc = __builtin_amdgcn_wmma_f32_16x16x4_f32(
            /*neg_a=*/false, a, /*neg_b=*/false, b,
            /*c_mod=*/(short)0, acc, /*reuse_a=*/false, /*reuse_b=*/false);
    }

    float* yout = Y + (size_t)(m0 + half * 8) * 16 + l;
#pragma unroll
    for (int r = 0; r < 8; ++r) yout[(size_t)r * 16] = acc[r];
}

// ---------------------------------------------------------------------------
// Edge scatter: agg[dst] += t[src] * dis[src]*dis[dst], 16 lanes per edge
// t (N x 16 fp32 = 6.4 MB) is L2-resident -> gathers/atomics hit L2.
// ---------------------------------------------------------------------------
__global__ void scatter_kernel(const int* __restrict__ src,
                               const int* __restrict__ dst,
                               const float* __restrict__ dis,
                               const float* __restrict__ t,
                               float* __restrict__ agg, int e) {
    int idx = blockIdx.x * blockDim.x + threadIdx.x;
    int ed  = idx >> 4;
    int f   = idx & 15;
    if (ed >= e) return;
    int s = src[ed], d = dst[ed];
    float w = dis[s] * dis[d];
    atomicAdd(&agg[(size_t)d * 16 + f], t[(size_t)s * 16 + f] * w);
}

// self-loop term + bias + relu
__global__ void finalize_kernel(const float* __restrict__ agg,
                                const float* __restrict__ t,
                                const float* __restrict__ dis,
                                const float* __restrict__ b,
                                float* __restrict__ h, int n) {
    int idx = blockIdx.x * blockDim.x + threadIdx.x;
    int i = idx >> 4, f = idx & 15;
    if (i >= n) return;
    float di = dis[i];
    float v  = agg[idx] + t[idx] * di * di + b[f];
    h[idx]   = fmaxf(v, 0.0f);
}

// ---------------------------------------------------------------------------
// Global max pool over nodes (values >= 0 after relu -> uint-bit max is exact)
// ---------------------------------------------------------------------------
__global__ void maxpool_kernel(const float* __restrict__ h, unsigned* pooled, int total) {
    __shared__ unsigned smax[16];
    if (threadIdx.x < 16) smax[threadIdx.x] = 0u;
    __syncthreads();
    int idx = blockIdx.x * blockDim.x + threadIdx.x;
    if (idx < total) atomicMax(&smax[idx & 15], __float_as_uint(h[idx]));
    __syncthreads();
    if (threadIdx.x < 16) atomicMax(&pooled[threadIdx.x], smax[threadIdx.x]);
}

// out[a] = sum_f pooled[f] * Wc[f,a] + bc[a]
__global__ void head_kernel(const unsigned* __restrict__ pooled,
                            const float* __restrict__ Wc,
                            const float* __restrict__ bc,
                            float* __restrict__ out, int A) {
    int a = threadIdx.x;
    if (a >= A) return;
    float acc = bc[a];
#pragma unroll
    for (int f = 0; f < 16; ++f)
        acc += __uint_as_float(pooled[f]) * Wc[f * A + a];
    out[a] = acc;
}

// ---------------------------------------------------------------------------
extern "C" void kernel_launch(void* const* d_in, const int* in_sizes, int n_in,
                              void* d_out, int out_size, void* d_ws, size_t ws_size,
                              hipStream_t stream) {
    const float* x    = (const float*)d_in[0];   // [N,128]
    const int*   ei   = (const int*)d_in[1];     // [2,E] : row0=src, row1=dst
    const float* W1   = (const float*)d_in[2];   // [128,16]
    const float* b1   = (const float*)d_in[3];   // [16]
    const float* W2   = (const float*)d_in[4];   // [16,16]
    const float* b2   = (const float*)d_in[5];   // [16]
    const float* Wc   = (const float*)d_in[6];   // [16,10]
    const float* bc   = (const float*)d_in[7];   // [10]
    float*       out  = (float*)d_out;

    const int N = in_sizes[0] / 128;
    const int E = in_sizes[1] / 2;
    const int A = out_size;
    const int F_IN = 128, H = 16;

    const int* e_src = ei;
    const int* e_dst = ei + E;

    // workspace layout (fp32)
    float* ws  = (float*)d_ws;
    float* dis = ws;                        // [N]  (deg then deg^-1/2 in place)
    float* t   = dis + (size_t)N;           // [N,16]  transformed features (per layer)
    float* agg = t   + (size_t)N * H;       // [N,16]  scatter accumulator
    float* h   = agg + (size_t)N * H;       // [N,16]  activations (per layer)
    unsigned* pooled = (unsigned*)(h + (size_t)N * H);  // [16]

    const int B = 256;
    auto blocks = [](long long work, int blk) { return (int)((work + blk - 1) / blk); };

    // --- degree normalization --------------------------------------------
    init_deg_kernel<<<blocks(N, B), B, 0, stream>>>(dis, N);
    count_deg_kernel<<<blocks(E, B), B, 0, stream>>>(e_dst, dis, E);
    rsqrt_kernel<<<blocks(N, B), B, 0, stream>>>(dis, N);

    const int tiles      = (N + 15) / 16;           // 6250
    const int gemm_grid  = blocks((long long)tiles * 32, B);
    const long long ef   = (long long)E * H;        // edge*feature work items
    const long long nf   = (long long)N * H;

    // --- layer 1 ----------------------------------------------------------
    gemm16_wmma_f32<<<gemm_grid, B, 0, stream>>>(x, W1, t, N, F_IN);
    hipMemsetAsync(agg, 0, (size_t)N * H * sizeof(float), stream);
    scatter_kernel<<<blocks(ef, B), B, 0, stream>>>(e_src, e_dst, dis, t, agg, E);
    finalize_kernel<<<blocks(nf, B), B, 0, stream>>>(agg, t, dis, b1, h, N);

    // --- layer 2 ----------------------------------------------------------
    gemm16_wmma_f32<<<gemm_grid, B, 0, stream>>>(h, W2, t, N, H);
    hipMemsetAsync(agg, 0, (size_t)N * H * sizeof(float), stream);
    scatter_kernel<<<blocks(ef, B), B, 0, stream>>>(e_src, e_dst, dis, t, agg, E);
    finalize_kernel<<<blocks(nf, B), B, 0, stream>>>(agg, t, dis, b2, h, N);

    // --- pool + head ------------------------------------------------------
    hipMemsetAsync(pooled, 0, 16 * sizeof(unsigned), stream);
    maxpool_kernel<<<blocks(nf, B), B, 0, stream>>>(h, pooled, (int)nf);
    head_kernel<<<1, 32, 0, stream>>>(pooled, Wc, bc, out, A);
}